// QuantumNeuralNet_10033043603487
// MI455X (gfx1250) — compile-verified
//
#include <hip/hip_runtime.h>
#include <hip/hip_bf16.h>
#include <math.h>

typedef float v2f __attribute__((ext_vector_type(2)));
typedef float v8f __attribute__((ext_vector_type(8)));

#define NFEAT   784
#define NF4     196      // 784/4
#define NSTATE  16
#define LAYERS  3
#define RBLK    128      // row-blocks for column-sum pass
#define BLK     256

// ---------------- Pass 1: deterministic partial column sums ----------------
__global__ __launch_bounds__(BLK) void colsum_kernel(const float* __restrict__ x,
                                                     float* __restrict__ partials,
                                                     int rowsPerBlock)
{
    int t = threadIdx.x;
    if (t >= NF4) return;
    const float4* xr = (const float4*)x;
    long long r0 = (long long)blockIdx.x * rowsPerBlock;
    float4 acc = make_float4(0.f, 0.f, 0.f, 0.f);
#pragma unroll 4
    for (int r = 0; r < rowsPerBlock; ++r) {
        float4 v = xr[(r0 + r) * NF4 + t];
        acc.x += v.x; acc.y += v.y; acc.z += v.z; acc.w += v.w;
    }
    float* p = partials + (long long)blockIdx.x * NFEAT + t * 4;
    p[0] = acc.x; p[1] = acc.y; p[2] = acc.z; p[3] = acc.w;
}

// ------ Pass 2: reduce partials, pick first 4 cols mean>0.5, gate trig ------
__global__ __launch_bounds__(BLK) void select_kernel(const float* __restrict__ partials,
                                                     const float* __restrict__ qw,
                                                     int nblk, float invB,
                                                     int* __restrict__ idxOut,
                                                     float* __restrict__ trigOut)
{
    __shared__ unsigned char flags[NFEAT];
    int t = threadIdx.x;

    // uniform layer-gate cos/sin (batch-invariant): trig[2k]=cos, trig[2k+1]=sin
    if (t < LAYERS * 4) {
        float th = 0.5f * qw[t];
        float s, c;
        __sincosf(th, &s, &c);
        trigOut[2 * t + 0] = c;
        trigOut[2 * t + 1] = s;
    }

    for (int c = t; c < NFEAT; c += BLK) {
        float s = 0.f;
        for (int k = 0; k < nblk; ++k) s += partials[(long long)k * NFEAT + c];
        flags[c] = (s * invB) > 0.5f ? 1 : 0;
    }
    __syncthreads();
    if (t == 0) {
        int id[4] = {0, 0, 0, 0};
        int n = 0;
        for (int c = 0; c < NFEAT && n < 4; ++c)
            if (flags[c]) id[n++] = c;
        idxOut[0] = id[0]; idxOut[1] = id[1]; idxOut[2] = id[2]; idxOut[3] = id[3];
    }
}

// ---------------- fast branch-free helpers ----------------------------------
__device__ __forceinline__ float fast_tanh(float x)
{
    // tanh(x) = 1 - 2*rcp(exp(2x)+1); v_exp_f32 + v_rcp_f32, branch-free,
    // safe at extremes: e->inf => 1-0 = 1 ; e->0 => 1-2 = -1
    float e = __expf(2.0f * x);
    return __builtin_fmaf(-2.0f, __builtin_amdgcn_rcpf(e + 1.0f), 1.0f);
}

// ---------------- 4-qubit statevector helpers (fully unrolled) --------------
__device__ __forceinline__ void rx_gate(float sr[NSTATE], float si[NSTATE],
                                        int stride, float c, float s)
{
#pragma unroll
    for (int i = 0; i < NSTATE; ++i) {
        if (i & stride) continue;
        int j = i | stride;
        float ar = sr[i], ai = si[i], br = sr[j], bi = si[j];
        // RX: a' = c*a - i*s*b ; b' = -i*s*a + c*b
        sr[i] = c * ar + s * bi;
        si[i] = c * ai - s * br;
        sr[j] = c * br + s * ai;
        si[j] = c * bi - s * ar;
    }
}

__device__ __forceinline__ void cnot_gate(float sr[NSTATE], float si[NSTATE],
                                          int sc, int st)
{
#pragma unroll
    for (int i = 0; i < NSTATE; ++i) {
        if (!(i & sc) || (i & st)) continue;
        int j = i | st;
        float tr = sr[i]; sr[i] = sr[j]; sr[j] = tr;
        float ti = si[i]; si[i] = si[j]; si[j] = ti;
    }
}

// ---------------- Pass 3: fused gather -> dense1 -> circuit -> dense2 -------
__global__ __launch_bounds__(BLK) void qnn_main_kernel(
    const float* __restrict__ x,
    const float* __restrict__ W1, const float* __restrict__ b1,
    const float* __restrict__ W2, const float* __restrict__ b2,
    const int* __restrict__ idx, const float* __restrict__ trig,
    float* __restrict__ out)
{
    __shared__ float s_xs[BLK][4];   // gathered features
    __shared__ float s_h [BLK][4];   // tanh dense1 output
    __shared__ float s_z [BLK][4];   // circuit expectation values
    __shared__ float s_out[BLK * 10];

    int tid  = threadIdx.x;
    int lane = tid & 31;
    int wbase = tid & ~31;           // this wave's sample base within block
    int half  = lane >> 4;           // 0: lanes 0-15, 1: lanes 16-31
    int lm    = lane & 15;
    long long sample = (long long)blockIdx.x * BLK + tid;

    // ---- gather the 4 selected features for this sample ----
    int i0 = idx[0], i1 = idx[1], i2 = idx[2], i3 = idx[3];
    const float* row = x + sample * NFEAT;
    s_xs[tid][0] = row[i0];
    s_xs[tid][1] = row[i1];
    s_xs[tid][2] = row[i2];
    s_xs[tid][3] = row[i3];
    __syncthreads();

    // ---- GEMM 1: preact = xs @ W1^T + b1  via V_WMMA_F32_16X16X4_F32 ----
    // A (16x4): lanes0-15 V0=K0,V1=K1; lanes16-31 V0=K2,V1=K3
    // B (4x16): VGPR v holds row K=v (lanes0-15) / K=v+2 (lanes16-31); B[k][n]=W1[n][k]
    {
        float bb0 = (lm < 4) ? W1[lm * 4 + (0 + 2 * half)] : 0.f;
        float bb1 = (lm < 4) ? W1[lm * 4 + (1 + 2 * half)] : 0.f;
        v2f Bm = {bb0, bb1};
        float cval = (lm < 4) ? b1[lm] : 0.f;
        v8f Cm;
#pragma unroll
        for (int v = 0; v < 8; ++v) Cm[v] = cval;
#pragma unroll
        for (int sub = 0; sub < 2; ++sub) {
            int m = wbase + sub * 16 + lm;            // A rows: 16 samples
            v2f Am = {s_xs[m][2 * half + 0], s_xs[m][2 * half + 1]};
            v8f D = __builtin_amdgcn_wmma_f32_16x16x4_f32(
                false, Am, false, Bm, (short)0, Cm, false, false);
            if (lm < 4) {
#pragma unroll
                for (int v = 0; v < 8; ++v) {
                    int mm = wbase + sub * 16 + v + 8 * half;
                    s_h[mm][lm] = fast_tanh(D[v]);
                }
            }
        }
    }
    __syncthreads();

    // ---- 4-qubit circuit, one sample per thread, all in registers ----
    {
        float sr[NSTATE], si[NSTATE];
#pragma unroll
        for (int i = 0; i < NSTATE; ++i) { sr[i] = 0.f; si[i] = 0.f; }
        sr[0] = 1.f;

        // AngleEmbedding: RX(h[w]) on wire w (wire w -> bit stride 1<<(3-w))
#pragma unroll
        for (int w = 0; w < 4; ++w) {
            float th = 0.5f * s_h[tid][w];
            float s, c;
            __sincosf(th, &s, &c);
            rx_gate(sr, si, 1 << (3 - w), c, s);
        }
        // BasicEntanglerLayers: RX(qw[l][w]) then CNOT ring (0,1)(1,2)(2,3)(3,0)
#pragma unroll
        for (int l = 0; l < LAYERS; ++l) {
#pragma unroll
            for (int w = 0; w < 4; ++w) {
                float c = trig[2 * (l * 4 + w) + 0];
                float s = trig[2 * (l * 4 + w) + 1];
                rx_gate(sr, si, 1 << (3 - w), c, s);
            }
            cnot_gate(sr, si, 8, 4);  // (0,1)
            cnot_gate(sr, si, 4, 2);  // (1,2)
            cnot_gate(sr, si, 2, 1);  // (2,3)
            cnot_gate(sr, si, 1, 8);  // (3,0)
        }
        // <Z_w> = sum_{bit_w=0} |amp|^2 - sum_{bit_w=1} |amp|^2
#pragma unroll
        for (int w = 0; w < 4; ++w) {
            int st = 1 << (3 - w);
            float acc = 0.f;
#pragma unroll
            for (int i = 0; i < NSTATE; ++i) {
                float p = sr[i] * sr[i] + si[i] * si[i];
                acc += (i & st) ? -p : p;
            }
            s_z[tid][w] = acc;
        }
    }
    __syncthreads();

    // ---- GEMM 2: out = z @ W2^T + b2  via V_WMMA_F32_16X16X4_F32 ----
    {
        float bb0 = (lm < 10) ? W2[lm * 4 + (0 + 2 * half)] : 0.f;
        float bb1 = (lm < 10) ? W2[lm * 4 + (1 + 2 * half)] : 0.f;
        v2f Bm = {bb0, bb1};
        float cval = (lm < 10) ? b2[lm] : 0.f;
        v8f Cm;
#pragma unroll
        for (int v = 0; v < 8; ++v) Cm[v] = cval;
#pragma unroll
        for (int sub = 0; sub < 2; ++sub) {
            int m = wbase + sub * 16 + lm;
            v2f Am = {s_z[m][2 * half + 0], s_z[m][2 * half + 1]};
            v8f D = __builtin_amdgcn_wmma_f32_16x16x4_f32(
                false, Am, false, Bm, (short)0, Cm, false, false);
            if (lm < 10) {
#pragma unroll
                for (int v = 0; v < 8; ++v) {
                    int mm = wbase + sub * 16 + v + 8 * half;
                    s_out[mm * 10 + lm] = D[v];
                }
            }
        }
    }
    __syncthreads();

    // ---- coalesced output store ----
    long long obase = (long long)blockIdx.x * BLK * 10;
    for (int i = tid; i < BLK * 10; i += BLK)
        out[obase + i] = s_out[i];
}

// ---------------------------------------------------------------------------
extern "C" void kernel_launch(void* const* d_in, const int* in_sizes, int n_in,
                              void* d_out, int out_size, void* d_ws, size_t ws_size,
                              hipStream_t stream)
{
    const float* x  = (const float*)d_in[0];   // [B,784]
    const float* qw = (const float*)d_in[1];   // [3,4]
    const float* W1 = (const float*)d_in[2];   // [4,4]
    const float* b1 = (const float*)d_in[3];   // [4]
    const float* W2 = (const float*)d_in[4];   // [10,4]
    const float* b2 = (const float*)d_in[5];   // [10]
    float* out = (float*)d_out;

    int B = in_sizes[0] / NFEAT;               // 65536
    int rowsPerBlock = B / RBLK;

    int*   idx      = (int*)d_ws;                       // ws+0   : 4 ints
    float* trig     = (float*)((char*)d_ws + 64);       // ws+64  : 24 floats
    float* partials = (float*)((char*)d_ws + 256);      // ws+256 : RBLK*784 floats

    colsum_kernel<<<RBLK, BLK, 0, stream>>>(x, partials, rowsPerBlock);
    select_kernel<<<1, BLK, 0, stream>>>(partials, qw, RBLK, 1.0f / (float)B, idx, trig);
    qnn_main_kernel<<<B / BLK, BLK, 0, stream>>>(x, W1, b1, W2, b2, idx, trig, out);
}